// GCNNet_26182120636977
// MI455X (gfx1250) — compile-verified
//
#include <hip/hip_runtime.h>
#include <math.h>

typedef float v2f __attribute__((ext_vector_type(2)));
typedef float v8f __attribute__((ext_vector_type(8)));

#define N_NODES   50000
#define N_EDGES   320000
#define IN_FEAT   128
#define HIDDEN    256
#define N_CLASSES 40
#define BN_EPS    1e-5f

// ---------------- degree / normalization ----------------
__global__ void k_deg_init(float* __restrict__ deg, int n) {
    int i = blockIdx.x * blockDim.x + threadIdx.x;
    if (i < n) deg[i] = 1.0f;   // self-loop contribution
}

__global__ void k_deg_edges(float* __restrict__ deg, const long long* __restrict__ dst, int e) {
    int i = blockIdx.x * blockDim.x + threadIdx.x;
    if (i < e) {
        int d = (int)dst[i];
        atomicAdd(&deg[d], 1.0f);
    }
}

__global__ void k_dinv(float* __restrict__ d, int n) {
    int i = blockIdx.x * blockDim.x + threadIdx.x;
    if (i < n) d[i] = rsqrtf(d[i]);   // deg >= 1 always (self loop)
}

// ---------------- fp32 WMMA GEMM: C[M x N] = A[M x K] * B[K x N] ----------------
// block = 128 threads (4 waves). wave computes 32x64. grid = (ceil(M/128), N/64).
// K multiple of 64, N multiple of 64.
#define KT 64
__global__ __launch_bounds__(128) void k_wmma_gemm(
    const float* __restrict__ A, const float* __restrict__ B, float* __restrict__ C,
    int M, int N, int K, int ldc)
{
    __shared__ float ldsB[KT * 64];   // K-pairs interleaved: [p][n][parity], p = k/2

    const int tid   = threadIdx.x;
    const int lane  = tid & 31;
    const int wave  = tid >> 5;
    const int hhalf = lane >> 4;        // 0: lanes 0-15, 1: lanes 16-31
    const int l16   = lane & 15;
    const int koff  = hhalf * 2;        // K sub-offset per WMMA f32 A/B layout

    const int rowBase = blockIdx.x * 128 + wave * 32;
    const int n0      = blockIdx.y * 64;

    int r0 = rowBase + l16;        if (r0 > M - 1) r0 = M - 1;
    int r1 = rowBase + 16 + l16;   if (r1 > M - 1) r1 = M - 1;
    const float* a0 = A + (size_t)r0 * K;
    const float* a1 = A + (size_t)r1 * K;

    v8f acc[2][4];
#pragma unroll
    for (int mt = 0; mt < 2; ++mt)
#pragma unroll
        for (int j = 0; j < 4; ++j)
#pragma unroll
            for (int i = 0; i < 8; ++i) acc[mt][j][i] = 0.0f;

    for (int k0 = 0; k0 < K; k0 += KT) {
        __syncthreads();
        // stage B tile: rows k0..k0+63, cols n0..n0+63, interleaved as (even,odd) k pairs
        for (int idx = tid; idx < KT * 64; idx += 128) {
            int kk = idx >> 6;
            int nn = idx & 63;
            float v = B[(size_t)(k0 + kk) * N + (n0 + nn)];
            ldsB[(kk >> 1) * 128 + nn * 2 + (kk & 1)] = v;
        }
        __syncthreads();

#pragma unroll 4
        for (int kk = 0; kk < KT; kk += 4) {
            const int p = (kk + koff) >> 1;
            v2f av0 = *(const v2f*)(a0 + k0 + kk + koff);
            v2f av1 = *(const v2f*)(a1 + k0 + kk + koff);
            const float* bbase = ldsB + p * 128;
#pragma unroll
            for (int j = 0; j < 4; ++j) {
                v2f bv = *(const v2f*)(bbase + (j * 16 + l16) * 2);
                acc[0][j] = __builtin_amdgcn_wmma_f32_16x16x4_f32(
                    false, av0, false, bv, (short)0, acc[0][j], false, false);
                acc[1][j] = __builtin_amdgcn_wmma_f32_16x16x4_f32(
                    false, av1, false, bv, (short)0, acc[1][j], false, false);
            }
        }
    }

    // store: C/D layout -> VGPR i holds row (i + 8*half), col = l16
#pragma unroll
    for (int mt = 0; mt < 2; ++mt)
#pragma unroll
        for (int j = 0; j < 4; ++j)
#pragma unroll
            for (int i = 0; i < 8; ++i) {
                int row = rowBase + mt * 16 + hhalf * 8 + i;
                if (row < M) C[(size_t)row * ldc + n0 + j * 16 + l16] = acc[mt][j][i];
            }
}

// ---------------- aggregation: out = D^-1/2 A D^-1/2 * H + bias ----------------
__global__ void k_agg_init(float* __restrict__ out, const float* __restrict__ H,
                           const float* __restrict__ dinv, const float* __restrict__ bias,
                           int F, int ldH, int ldO)
{
    int node = blockIdx.x;
    float w = dinv[node]; w *= w;       // self-loop weight
    for (int f = threadIdx.x; f < F; f += blockDim.x)
        out[(size_t)node * ldO + f] = w * H[(size_t)node * ldH + f] + bias[f];
}

__global__ void k_agg_edges(float* __restrict__ out, const float* __restrict__ H,
                            const float* __restrict__ dinv,
                            const long long* __restrict__ src, const long long* __restrict__ dst,
                            int F, int ldH, int ldO)
{
    int e = blockIdx.x;
    int s = (int)src[e];
    int d = (int)dst[e];
    float w = dinv[s] * dinv[d];
    for (int f = threadIdx.x; f < F; f += blockDim.x)
        atomicAdd(&out[(size_t)d * ldO + f], w * H[(size_t)s * ldH + f]);
}

// ---------------- batchnorm (training-mode stats) + ReLU ----------------
__global__ void k_zero(float* __restrict__ p, int n) {
    int i = blockIdx.x * blockDim.x + threadIdx.x;
    if (i < n) p[i] = 0.0f;
}

__global__ void k_bn_stats(const float* __restrict__ X, float* __restrict__ stats,
                           int nodes, int F, int rowsPerBlock)
{
    int f  = threadIdx.x;               // blockDim.x == F
    int r0 = blockIdx.x * rowsPerBlock;
    int r1 = r0 + rowsPerBlock; if (r1 > nodes) r1 = nodes;
    float s = 0.0f, s2 = 0.0f;
    for (int r = r0; r < r1; ++r) {
        float v = X[(size_t)r * F + f];
        s += v; s2 += v * v;
    }
    atomicAdd(&stats[f], s);
    atomicAdd(&stats[F + f], s2);
}

__global__ void k_bn_apply(float* __restrict__ X, const float* __restrict__ stats,
                           const float* __restrict__ g, const float* __restrict__ be,
                           int nodes, int F)
{
    int f = threadIdx.x;
    int node = blockIdx.x;
    float invN = 1.0f / (float)nodes;
    float mean = stats[f] * invN;
    float var  = stats[F + f] * invN - mean * mean;
    float sc   = g[f] * rsqrtf(var + BN_EPS);
    size_t idx = (size_t)node * F + f;
    float y = (X[idx] - mean) * sc + be[f];
    X[idx] = fmaxf(y, 0.0f);
}

// ---------------- W3 padding (256x40 -> 256x64) ----------------
__global__ void k_padw3(float* __restrict__ Wp, const float* __restrict__ W) {
    int i = blockIdx.x * blockDim.x + threadIdx.x;
    if (i < HIDDEN * 64) {
        int k = i >> 6, c = i & 63;
        Wp[i] = (c < N_CLASSES) ? W[k * N_CLASSES + c] : 0.0f;
    }
}

// ---------------- row log_softmax (wave32 per row) ----------------
__global__ void k_logsoftmax(float* __restrict__ X, int nodes, int C) {
    int lane = threadIdx.x & 31;
    int row  = blockIdx.x * (blockDim.x >> 5) + (threadIdx.x >> 5);
    if (row >= nodes) return;
    float* p = X + (size_t)row * C;
    int f1 = lane + 32;
    float v0 = p[lane];                                   // lane < 32 <= C
    float v1 = (f1 < C) ? p[f1] : -__builtin_inff();
    float m = fmaxf(v0, v1);
#pragma unroll
    for (int o = 16; o > 0; o >>= 1) m = fmaxf(m, __shfl_xor(m, o, 32));
    float e = expf(v0 - m) + ((f1 < C) ? expf(v1 - m) : 0.0f);
#pragma unroll
    for (int o = 16; o > 0; o >>= 1) e += __shfl_xor(e, o, 32);
    float ls = logf(e);
    p[lane] = v0 - m - ls;
    if (f1 < C) p[f1] = v1 - m - ls;
}

// ---------------- driver ----------------
extern "C" void kernel_launch(void* const* d_in, const int* in_sizes, int n_in,
                              void* d_out, int out_size, void* d_ws, size_t ws_size,
                              hipStream_t stream)
{
    (void)in_sizes; (void)n_in; (void)out_size; (void)ws_size;
    const float*     x   = (const float*)d_in[0];
    const long long* ei  = (const long long*)d_in[1];   // int64 [2, E]
    const float*     W1  = (const float*)d_in[2];
    const float*     b1  = (const float*)d_in[3];
    const float*     g1  = (const float*)d_in[4];
    const float*     be1 = (const float*)d_in[5];
    const float*     W2  = (const float*)d_in[6];
    const float*     b2  = (const float*)d_in[7];
    const float*     g2  = (const float*)d_in[8];
    const float*     be2 = (const float*)d_in[9];
    const float*     W3  = (const float*)d_in[10];
    const float*     b3  = (const float*)d_in[11];
    float* out = (float*)d_out;

    const long long* src = ei;
    const long long* dst = ei + N_EDGES;

    // workspace carve-up (floats)
    float* wsf   = (float*)d_ws;
    float* dinv  = wsf;                                   // 50000
    float* P     = wsf + 50016;                           // 50000*256 (H buffer)
    float* Q     = P + (size_t)N_NODES * HIDDEN;          // 50000*256 (A buffer)
    float* W3p   = Q + (size_t)N_NODES * HIDDEN;          // 256*64
    float* stats = W3p + HIDDEN * 64;                     // 512

    // --- normalization coefficients ---
    k_deg_init <<<(N_NODES + 255) / 256, 256, 0, stream>>>(dinv, N_NODES);
    k_deg_edges<<<(N_EDGES + 255) / 256, 256, 0, stream>>>(dinv, dst, N_EDGES);
    k_dinv     <<<(N_NODES + 255) / 256, 256, 0, stream>>>(dinv, N_NODES);

    dim3 gemmGrid((N_NODES + 127) / 128, HIDDEN / 64);

    // --- layer 1: H1 = X @ W1 ; A1 = norm-agg(H1) + b1 ; BN+ReLU ---
    k_wmma_gemm<<<gemmGrid, 128, 0, stream>>>(x, W1, P, N_NODES, HIDDEN, IN_FEAT, HIDDEN);
    k_agg_init <<<N_NODES, 256, 0, stream>>>(Q, P, dinv, b1, HIDDEN, HIDDEN, HIDDEN);
    k_agg_edges<<<N_EDGES, 256, 0, stream>>>(Q, P, dinv, src, dst, HIDDEN, HIDDEN, HIDDEN);
    k_zero     <<<2, 256, 0, stream>>>(stats, 2 * HIDDEN);
    k_bn_stats <<<(N_NODES + 127) / 128, HIDDEN, 0, stream>>>(Q, stats, N_NODES, HIDDEN, 128);
    k_bn_apply <<<N_NODES, HIDDEN, 0, stream>>>(Q, stats, g1, be1, N_NODES, HIDDEN);

    // --- layer 2 ---
    k_wmma_gemm<<<gemmGrid, 128, 0, stream>>>(Q, W2, P, N_NODES, HIDDEN, HIDDEN, HIDDEN);
    k_agg_init <<<N_NODES, 256, 0, stream>>>(Q, P, dinv, b2, HIDDEN, HIDDEN, HIDDEN);
    k_agg_edges<<<N_EDGES, 256, 0, stream>>>(Q, P, dinv, src, dst, HIDDEN, HIDDEN, HIDDEN);
    k_zero     <<<2, 256, 0, stream>>>(stats, 2 * HIDDEN);
    k_bn_stats <<<(N_NODES + 127) / 128, HIDDEN, 0, stream>>>(Q, stats, N_NODES, HIDDEN, 128);
    k_bn_apply <<<N_NODES, HIDDEN, 0, stream>>>(Q, stats, g2, be2, N_NODES, HIDDEN);

    // --- layer 3 (N padded 40 -> 64 for WMMA) + log_softmax ---
    k_padw3<<<(HIDDEN * 64 + 255) / 256, 256, 0, stream>>>(W3p, W3);
    dim3 gemmGrid3((N_NODES + 127) / 128, 1);
    k_wmma_gemm<<<gemmGrid3, 128, 0, stream>>>(Q, W3p, P, N_NODES, 64, HIDDEN, 64);
    k_agg_init <<<N_NODES, 64, 0, stream>>>(out, P, dinv, b3, N_CLASSES, 64, N_CLASSES);
    k_agg_edges<<<N_EDGES, 64, 0, stream>>>(out, P, dinv, src, dst, N_CLASSES, 64, N_CLASSES);
    k_logsoftmax<<<(N_NODES + 7) / 8, 256, 0, stream>>>(out, N_NODES, N_CLASSES);
}